// VGPMMs_46188078301711
// MI455X (gfx1250) — compile-verified
//
#include <hip/hip_runtime.h>
#include <hip/hip_bf16.h>
#include <math.h>

// ---------------- problem constants ----------------
#define Bb 8
#define Dd 129
#define Nn 9216
#define Kk 4
#define DP 144            // D padded to 9 tiles of 16
#define BK (Bb*Kk)        // 32 (b,k) pairs
#define LOG2PI_F 1.8378770664093453f

typedef __attribute__((ext_vector_type(2))) float v2f;
typedef __attribute__((ext_vector_type(8))) float v8f;
typedef unsigned int u32;
typedef unsigned long long u64;
typedef __attribute__((ext_vector_type(4))) unsigned int v4u;
typedef __attribute__((ext_vector_type(8))) int v8i;
typedef __attribute__((ext_vector_type(4))) int v4i;

// ============================================================
// TDM helper: 2D tile load Global -> LDS (Tensor Data Mover).
// D# per CDNA5 ISA 8.3/8.4: group0 = {flags, lds_addr, global_addr, type},
// group1 = {ctrl, tensor_dim0/1, tile_dim0/1/2, strides}. Groups 2/3 = 0 (2D).
// pad_code/amt_code: LDS pitch padding (interval 2^(c+1) DW, amount amt+1 DW).
// Toolchain here is the 6-arg clang-23 form:
//   (uint32x4, int32x8, int32x4, int32x4, int32x8, i32 cpol)
// ============================================================
__device__ __forceinline__ void tdm_load_2d(u32 lds_off, const void* gptr,
                                            u32 tile_x, u32 tile_y,
                                            u32 dim_x, u32 dim_y,
                                            u32 stride_x,
                                            u32 pad_en, u32 pad_code, u32 amt_code) {
    u64 ga = (u64)(size_t)gptr;
    v4u g0;
    g0.x = 1u;                                          // count=1, user desc
    g0.y = lds_off;                                     // lds_addr (bytes)
    g0.z = (u32)ga;                                     // global_addr[31:0]
    g0.w = (u32)((ga >> 32) & 0x01FFFFFFu) | (2u << 30);// addr[56:32] | type=2
    u32 w0 = (2u << 16) | (pad_en << 20) | (pad_code << 22) | (amt_code << 25);
    u32 w1 = (dim_x & 0xFFFFu) << 16;                   // tensor_dim0 lo16
    u32 w2 = ((dim_x >> 16) & 0xFFFFu) | ((dim_y & 0xFFFFu) << 16);
    u32 w3 = ((dim_y >> 16) & 0xFFFFu) | ((tile_x & 0xFFFFu) << 16);
    u32 w4 = (tile_y & 0xFFFFu);                        // tile_dim1 (tile_dim2=0)
    u32 w5 = stride_x;                                  // tensor_dim0_stride lo32
    v8i g1; g1[0]=(int)w0; g1[1]=(int)w1; g1[2]=(int)w2; g1[3]=(int)w3;
            g1[4]=(int)w4; g1[5]=(int)w5; g1[6]=0; g1[7]=0;
    v4i gz4 = {0,0,0,0};
    v8i gz8 = {0,0,0,0,0,0,0,0};
    __builtin_amdgcn_tensor_load_to_lds(g0, g1, gz4, gz4, gz8, 0);
}

// ============================================================
// K0: zero scalar accumulators
// ============================================================
__global__ void k_init(float* scal) {
    if (threadIdx.x < 8) scal[threadIdx.x] = 0.f;
}

// ============================================================
// K1: sg[bk] = sum_n gamma[bk][n]
// ============================================================
__global__ void k_sg(const float* __restrict__ gamma, float* __restrict__ sg) {
    __shared__ float red[256];
    int bk = blockIdx.x;
    float s = 0.f;
    for (int n = threadIdx.x; n < Nn; n += 256)
        s += gamma[(size_t)bk * Nn + n];
    red[threadIdx.x] = s; __syncthreads();
    for (int w = 128; w >= 1; w >>= 1) {
        if (threadIdx.x < w) red[threadIdx.x] += red[threadIdx.x + w];
        __syncthreads();
    }
    if (threadIdx.x == 0) sg[bk] = red[0];
}

// ============================================================
// K2: mu[bk][d] = (sum_n gamma[bk][n]*z[b][d][n]) / sg[bk]   (padded rows = 0)
// ============================================================
__global__ void k_mu(const float* __restrict__ z, const float* __restrict__ gamma,
                     const float* __restrict__ sg, float* __restrict__ mu) {
    int wave = threadIdx.x >> 5, lane = threadIdx.x & 31;
    int bk = blockIdx.x / 18;
    int d  = (blockIdx.x % 18) * 8 + wave;
    int b  = bk / Kk;
    float s = 0.f;
    if (d < Dd) {
        const float* zr = z + ((size_t)b * Dd + d) * Nn;
        const float* gr = gamma + (size_t)bk * Nn;
        for (int n = lane; n < Nn; n += 32) s += gr[n] * zr[n];
    }
    for (int m = 16; m >= 1; m >>= 1) s += __shfl_xor(s, m, 32);
    if (lane == 0) mu[bk * DP + d] = (d < Dd) ? s / sg[bk] : 0.f;
}

// ============================================================
// K3: covariance via fp32 WMMA 16x16x4, TDM-fed & double-buffered.
// cov[d,e] = (1/sg) * sum_n gamma[n]*(z[d,n]-mu[d])*(z[e,n]-mu[e])
// grid = BK*81 blocks (one 16x16 tile), 32 threads (1 wave -> no barriers).
// gamma row (36KB) DMA'd once; per-32-col chunk two 16x32 strips DMA'd with
// LDS pitch 34 (TDM pad: 32 DW interval + 2 DW) while WMMA consumes the
// previous chunk. Sync via s_wait_tensorcnt (TDM completes in-order per wave).
// ============================================================
__global__ void __launch_bounds__(32) k_cov(const float* __restrict__ z,
                                            const float* __restrict__ gamma,
                                            const float* __restrict__ mu,
                                            const float* __restrict__ sg,
                                            float* __restrict__ covL) {
    __shared__ float gs[Nn];           // whole gamma row: 36 KB
    __shared__ float As[2][16 * 34];   // double-buffered strips, pitch 34
    __shared__ float Bs[2][16 * 34];
    int tile = blockIdx.x % 81;
    int bk   = blockIdx.x / 81;
    int b    = bk / Kk;
    int d0 = (tile / 9) * 16, e0 = (tile % 9) * 16;
    int lane = threadIdx.x;
    int L = lane & 15, hi = lane >> 4;
    const float* zb = z + (size_t)b * Dd * Nn;
    float mud = mu[bk * DP + d0 + L];  // pad entries are 0
    float mue = mu[bk * DP + e0 + L];
    int rowsA = (Dd - d0 < 16) ? (Dd - d0) : 16;
    int rowsB = (Dd - e0 < 16) ? (Dd - e0) : 16;

    // zero only the clipped pad rows (disjoint from TDM writes -> no race)
    for (int t = lane; t < 16 * 34; t += 32) {
        int r = t / 34;
        if (r >= rowsA) { As[0][t] = 0.f; As[1][t] = 0.f; }
        if (r >= rowsB) { Bs[0][t] = 0.f; Bs[1][t] = 0.f; }
    }

    // gamma row DMA (1D: 9216 x 1, no pad)
    tdm_load_2d((u32)(size_t)gs, gamma + (size_t)bk * Nn,
                Nn, 1, Nn, 1, Nn, 0, 0, 0);

    // strip DMA for chunk c into buffer c&1
    auto issue = [&](int c) {
        int n0 = c * 32;
        tdm_load_2d((u32)(size_t)As[c & 1], zb + (size_t)d0 * Nn + n0,
                    32, (u32)rowsA, 32, (u32)rowsA, Nn, 1, 4, 1); // pitch 34
        tdm_load_2d((u32)(size_t)Bs[c & 1], zb + (size_t)e0 * Nn + n0,
                    32, (u32)rowsB, 32, (u32)rowsB, Nn, 1, 4, 1);
    };

    v8f c8 = {};
    issue(0);
    for (int c = 0; c < Nn / 32; ++c) {
        if (c + 1 < Nn / 32) {
            issue(c + 1);
            __builtin_amdgcn_s_wait_tensorcnt(2);  // chunk c (and gamma) resident
        } else {
            __builtin_amdgcn_s_wait_tensorcnt(0);
        }
        __asm__ volatile("" ::: "memory");
        const float* Ab = As[c & 1];
        const float* Bf = Bs[c & 1];
        int n0 = c * 32;
        #pragma unroll
        for (int step = 0; step < 8; ++step) {
            int kb = step * 4 + 2 * hi;
            v2f a, bb;
            a.x  = gs[n0 + kb]     * (Ab[L * 34 + kb]     - mud);
            a.y  = gs[n0 + kb + 1] * (Ab[L * 34 + kb + 1] - mud);
            bb.x = Bf[L * 34 + kb]     - mue;
            bb.y = Bf[L * 34 + kb + 1] - mue;
            c8 = __builtin_amdgcn_wmma_f32_16x16x4_f32(false, a, false, bb,
                                                       (short)0, c8, false, false);
        }
    }
    float inv = 1.f / sg[bk];
    float* C = covL + (size_t)bk * DP * DP;
    #pragma unroll
    for (int r = 0; r < 8; r++) {
        int d = d0 + r + 8 * hi;
        C[(size_t)d * DP + e0 + L] = c8[r] * inv;
    }
}

// ============================================================
// K4: make_pd + in-place left-looking Cholesky + logdet (first 129 diag)
// ============================================================
__global__ void k_chol(float* __restrict__ covL, float* __restrict__ logdet) {
    __shared__ float shLjj;
    __shared__ float red[256];
    int bk = blockIdx.x;
    float* A = covL + (size_t)bk * DP * DP;
    int tid = threadIdx.x;
    if (tid < DP) {
        float v = A[tid * DP + tid];
        float c = (v != v) ? 1e-6f : fmaxf(v, 1e-6f);
        A[tid * DP + tid] = v + c;
    }
    __syncthreads();
    for (int j = 0; j < DP; ++j) {
        int i = j + tid;
        float s = 0.f;
        if (i < DP) {
            s = A[(size_t)i * DP + j];
            const float* Ri = A + (size_t)i * DP;
            const float* Rj = A + (size_t)j * DP;
            for (int p = 0; p < j; ++p) s -= Ri[p] * Rj[p];
        }
        if (tid == 0) shLjj = sqrtf(s);
        __syncthreads();
        if (i < DP) A[(size_t)i * DP + j] = (tid == 0) ? shLjj : s / shLjj;
        __syncthreads();
    }
    for (int t = tid; t < DP * DP; t += 256) {
        int i = t / DP, j = t % DP;
        if (j > i) A[t] = 0.f;
    }
    red[tid] = (tid < Dd) ? logf(A[tid * DP + tid]) : 0.f;
    __syncthreads();
    for (int w = 128; w >= 1; w >>= 1) {
        if (tid < w) red[tid] += red[tid + w];
        __syncthreads();
    }
    if (tid == 0) logdet[bk] = red[0];
}

// ============================================================
// K5: Linv = L^-1 (lower triangular inverse), one column per thread
// ============================================================
__global__ void k_linv(const float* __restrict__ covL, float* __restrict__ Li) {
    __shared__ float Lp[DP * (DP + 1) / 2];
    int bk = blockIdx.x;
    const float* A = covL + (size_t)bk * DP * DP;
    for (int t = threadIdx.x; t < DP * DP; t += 256) {
        int i = t / DP, j = t % DP;
        if (j <= i) Lp[(i * (i + 1)) / 2 + j] = A[t];
    }
    __syncthreads();
    int c = threadIdx.x;
    if (c < DP) {
        float x[DP];
        for (int i = 0; i < DP; ++i) {
            if (i < c) { x[i] = 0.f; continue; }
            float s = (i == c) ? 1.f : 0.f;
            const float* Lr = &Lp[(i * (i + 1)) / 2];
            for (int p = c; p < i; ++p) s -= Lr[p] * x[p];
            x[i] = s / Lr[i];
        }
        float* out = Li + (size_t)bk * DP * DP;
        for (int i = 0; i < DP; ++i) out[(size_t)i * DP + c] = x[i];
    }
}

// ============================================================
// K6: maha[bk][n] = || Linv @ (z - mu) ||^2 via TDM + fp32 WMMA.
// grid = BK * (Nn/64) blocks, 128 threads (4 waves).
// One 129x64 TDM per block (pitch 66 via pad: 64 DW + 2 DW), then in-place
// mu subtraction, then 9 row-tiles x 36 K-steps of WMMA per wave.
// ============================================================
__global__ void __launch_bounds__(128) k_maha(const float* __restrict__ z,
                                              const float* __restrict__ mu,
                                              const float* __restrict__ Li,
                                              float* __restrict__ maha) {
    __shared__ float df[DP * 66];
    int chunks = Nn / 64;
    int bk = blockIdx.x / chunks;
    int n0 = (blockIdx.x % chunks) * 64;
    int b  = bk / Kk;
    const float* zb   = z + (size_t)b * Dd * Nn;
    const float* mubk = mu + bk * DP;

    // zero pad rows 129..143 (disjoint from TDM target rows)
    for (int t = threadIdx.x; t < (DP - Dd) * 66; t += 128)
        df[Dd * 66 + t] = 0.f;
    // wave 0 issues the tile DMA and waits for it
    if (threadIdx.x < 32) {
        tdm_load_2d((u32)(size_t)df, zb + n0,
                    64, Dd, 64, Dd, Nn, 1, 5, 1);   // pitch 66
        __builtin_amdgcn_s_wait_tensorcnt(0);
    }
    __syncthreads();
    // diff = z - mu (rows < 129)
    for (int t = threadIdx.x; t < Dd * 64; t += 128) {
        int r = t >> 6, c = t & 63;
        df[r * 66 + c] -= mubk[r];
    }
    __syncthreads();

    int wave = threadIdx.x >> 5, lane = threadIdx.x & 31;
    int L = lane & 15, hi = lane >> 4;
    int cw = wave * 16;
    const float* Libk = Li + (size_t)bk * DP * DP;
    float msum = 0.f;
    for (int rt = 0; rt < 9; ++rt) {
        v8f c8 = {};
        const float2* A2 = (const float2*)(Libk + (size_t)(rt * 16 + L) * DP);
        for (int kk = 0; kk < DP; kk += 4) {
            float2 av = A2[(kk >> 1) + hi];
            v2f a;  a.x = av.x; a.y = av.y;
            int kb = kk + 2 * hi;
            v2f bb; bb.x = df[kb * 66 + cw + L];
                    bb.y = df[(kb + 1) * 66 + cw + L];
            c8 = __builtin_amdgcn_wmma_f32_16x16x4_f32(false, a, false, bb,
                                                       (short)0, c8, false, false);
        }
        float s = 0.f;
        #pragma unroll
        for (int v = 0; v < 8; v++) s += c8[v] * c8[v];
        msum += s;
    }
    msum += __shfl_xor(msum, 16, 32);
    if (lane < 16) maha[(size_t)bk * Nn + n0 + cw + L] = msum;
}

// ============================================================
// K7: logsumexp over K + accumulate sum for enc_energy
// ============================================================
__global__ void k_lse(const float* __restrict__ maha, const float* __restrict__ sg,
                      const float* __restrict__ ld, float* __restrict__ acc) {
    __shared__ float red[256];
    int t = blockIdx.x * 256 + threadIdx.x;
    int b = t / Nn, n = t % Nn;
    float lp[Kk];
    float m = -1e30f;
    #pragma unroll
    for (int k = 0; k < Kk; ++k) {
        int bk = b * Kk + k;
        float v = -0.5f * ((float)Dd * LOG2PI_F + maha[(size_t)bk * Nn + n])
                  - ld[bk] + logf(sg[bk] / (float)Nn);
        lp[k] = v; m = fmaxf(m, v);
    }
    float s = 0.f;
    #pragma unroll
    for (int k = 0; k < Kk; ++k) s += expf(lp[k] - m);
    red[threadIdx.x] = m + logf(s);
    __syncthreads();
    for (int w = 128; w >= 1; w >>= 1) {
        if (threadIdx.x < w) red[threadIdx.x] += red[threadIdx.x + w];
        __syncthreads();
    }
    if (threadIdx.x == 0) atomicAdd(acc, red[0]);
}

// ============================================================
// K8: sampling: fs[b,s,:] = mu[b,comp] + L[b,comp] @ eps[s,b,:]
// ============================================================
__global__ void k_sample(const float* __restrict__ mu, const float* __restrict__ covL,
                         const int* __restrict__ comp, const float* __restrict__ eps,
                         float* __restrict__ out) {
    int sb = blockIdx.x;
    int s = sb / Bb, b = sb % Bb;
    int c = comp[s * Bb + b];
    int bk = b * Kk + c;
    int i = threadIdx.x;
    if (i < Dd) {
        const float* Lr = covL + (size_t)bk * DP * DP + (size_t)i * DP;
        const float* e  = eps + ((size_t)s * Bb + b) * Dd;
        float acc = mu[bk * DP + i];
        for (int j = 0; j <= i; ++j) acc += Lr[j] * e[j];
        out[((size_t)(b * Kk + s)) * Dd + i] = acc;   // [b,K,D] ravel
    }
}

// ============================================================
// K9: per-(b, sample, component) log prob of fs_fg under fg mixture
// ============================================================
__global__ void k_lpsamp(const float* __restrict__ fs, const float* __restrict__ mu,
                         const float* __restrict__ Li, const float* __restrict__ sg,
                         const float* __restrict__ ld, float* __restrict__ lpcomp) {
    __shared__ float dp[Dd];
    __shared__ float red[256];
    int id = blockIdx.x;                 // ((b*K + s)*K + kc)
    int kc = id % Kk; int bs = id / Kk;
    int s = bs % Kk, b = bs / Kk;
    int bk = b * Kk + kc;
    int tid = threadIdx.x;
    if (tid < Dd) dp[tid] = fs[((size_t)(b * Kk + s)) * Dd + tid] - mu[bk * DP + tid];
    __syncthreads();
    float y2 = 0.f;
    if (tid < Dd) {
        const float* Lr = Li + (size_t)bk * DP * DP + (size_t)tid * DP;
        float y = 0.f;
        for (int p = 0; p <= tid; ++p) y += Lr[p] * dp[p];
        y2 = y * y;
    }
    red[tid] = y2; __syncthreads();
    for (int w = 128; w >= 1; w >>= 1) {
        if (tid < w) red[tid] += red[tid + w];
        __syncthreads();
    }
    if (tid == 0)
        lpcomp[id] = -0.5f * ((float)Dd * LOG2PI_F + red[0]) - ld[bk]
                     + logf(sg[bk] / (float)Nn);
}

// ============================================================
// K10: KL(N_q || N_fg) per (b,k), accumulate sum
// ============================================================
__global__ void k_kl(const float* __restrict__ Lifg, const float* __restrict__ Lq,
                     const float* __restrict__ mufg, const float* __restrict__ muq,
                     const float* __restrict__ ldfg, const float* __restrict__ ldq,
                     float* __restrict__ klsum) {
    __shared__ float dm[Dd];
    __shared__ float red[256];
    int bk = blockIdx.x;
    int tid = threadIdx.x;
    if (tid < Dd) dm[tid] = mufg[bk * DP + tid] - muq[bk * DP + tid];
    __syncthreads();
    const float* Li = Lifg + (size_t)bk * DP * DP;
    const float* L0 = Lq + (size_t)bk * DP * DP;
    float acc = 0.f;                      // tr(M M^T), M = Linv_fg @ L_q (lower)
    for (int t = tid; t < Dd * Dd; t += 256) {
        int i = t / Dd, j = t % Dd;
        if (j <= i) {
            float m = 0.f;
            for (int p = j; p <= i; ++p) m += Li[(size_t)i * DP + p] * L0[(size_t)p * DP + j];
            acc += m * m;
        }
    }
    if (tid < Dd) {                       // maha term
        float y = 0.f;
        for (int p = 0; p <= tid; ++p) y += Li[(size_t)tid * DP + p] * dm[p];
        acc += y * y;
    }
    red[tid] = acc; __syncthreads();
    for (int w = 128; w >= 1; w >>= 1) {
        if (tid < w) red[tid] += red[tid + w];
        __syncthreads();
    }
    if (tid == 0) {
        float kl = 0.5f * (red[0] - (float)Dd) + (ldfg[bk] - ldq[bk]);
        atomicAdd(klsum, kl);
    }
}

// ============================================================
// K11: final scalars
// ============================================================
__global__ void k_final(const float* __restrict__ lpcomp, const float* __restrict__ scal,
                        float* __restrict__ out) {
    int lane = threadIdx.x;
    float lse = 0.f;
    if (lane < Bb * Kk) {
        float m = -1e30f, v[Kk];
        #pragma unroll
        for (int kc = 0; kc < Kk; ++kc) { v[kc] = lpcomp[lane * Kk + kc]; m = fmaxf(m, v[kc]); }
        float s = 0.f;
        #pragma unroll
        for (int kc = 0; kc < Kk; ++kc) s += expf(v[kc] - m);
        lse = m + logf(s);
    }
    for (int m = 16; m >= 1; m >>= 1) lse += __shfl_xor(lse, m, 32);
    if (lane == 0) {
        float enc_fg = -scal[0] / (float)(Bb * Nn);
        float enc_q  = -scal[1] / (float)(Bb * Nn);
        float proj = 1e-4f * (enc_fg + enc_q);
        float klm  = scal[2] / (float)(Bb * Kk);
        float lpm  = lse / (float)(Bb * Kk);
        out[2 * Bb * Kk * Dd]     = proj;
        out[2 * Bb * Kk * Dd + 1] = 1e-2f * (klm - lpm);
    }
}

// ============================================================
extern "C" void kernel_launch(void* const* d_in, const int* in_sizes, int n_in,
                              void* d_out, int out_size, void* d_ws, size_t ws_size,
                              hipStream_t stream) {
    const float* z[2]     = {(const float*)d_in[0], (const float*)d_in[1]};
    const float* gamma[2] = {(const float*)d_in[2], (const float*)d_in[3]};
    const float* eps[2]   = {(const float*)d_in[4], (const float*)d_in[5]};
    const int*   comp[2]  = {(const int*)d_in[6],   (const int*)d_in[7]};
    float* out = (float*)d_out;

    // workspace carve-out (floats)
    float* f = (float*)d_ws;
    size_t o = 0;
    auto alloc = [&](size_t n) { float* p = f + o; o += n; return p; };
    float* sg[2]   = {alloc(BK), alloc(BK)};
    float* mu[2]   = {alloc(BK * DP), alloc(BK * DP)};
    float* covL[2] = {alloc((size_t)BK * DP * DP), alloc((size_t)BK * DP * DP)};
    float* linv[2] = {alloc((size_t)BK * DP * DP), alloc((size_t)BK * DP * DP)};
    float* ld[2]   = {alloc(BK), alloc(BK)};
    float* mah[2]  = {alloc((size_t)BK * Nn), alloc((size_t)BK * Nn)};
    float* lpcomp  = alloc(Bb * Kk * Kk);
    float* scal    = alloc(8);   // [0]=enc_sum_fg [1]=enc_sum_q [2]=kl_sum

    k_init<<<1, 32, 0, stream>>>(scal);

    for (int s = 0; s < 2; ++s) {
        k_sg  <<<BK,       256, 0, stream>>>(gamma[s], sg[s]);
        k_mu  <<<BK * 18,  256, 0, stream>>>(z[s], gamma[s], sg[s], mu[s]);
        k_cov <<<BK * 81,   32, 0, stream>>>(z[s], gamma[s], mu[s], sg[s], covL[s]);
        k_chol<<<BK,       256, 0, stream>>>(covL[s], ld[s]);
        k_linv<<<BK,       256, 0, stream>>>(covL[s], linv[s]);
        k_maha<<<BK * (Nn / 64), 128, 0, stream>>>(z[s], mu[s], linv[s], mah[s]);
        k_lse <<<(Bb * Nn) / 256, 256, 0, stream>>>(mah[s], sg[s], ld[s], &scal[s]);
        k_sample<<<Kk * Bb, 160, 0, stream>>>(mu[s], covL[s], comp[s], eps[s],
                                              out + (size_t)s * Bb * Kk * Dd);
    }

    k_lpsamp<<<Bb * Kk * Kk, 256, 0, stream>>>(out, mu[0], linv[0], sg[0], ld[0], lpcomp);
    k_kl<<<BK, 256, 0, stream>>>(linv[0], covL[1], mu[0], mu[1], ld[0], ld[1], &scal[2]);
    k_final<<<1, 32, 0, stream>>>(lpcomp, scal, out);
}